// PointNetSetAbstraction_15324443312216
// MI455X (gfx1250) — compile-verified
//
#include <hip/hip_runtime.h>
#include <cfloat>
#include <stddef.h>

// ---------------- problem constants ----------------
#define BATCH   16
#define NPTS    8192
#define CFEAT   64
#define NPOINTS 512
#define NSAMP   32
#define KFEAT   (3 + CFEAT)                 // 67
#define PCOLS   (BATCH * NSAMP * KFEAT)     // 34304 = 268 * 128
#define BNEPS   1e-5f
#define PPT     (NPTS / 256)                // 32 points per thread

typedef float v2f __attribute__((ext_vector_type(2)));
typedef float v8f __attribute__((ext_vector_type(8)));

// =====================================================================
// Stage 1: farthest point sampling.  One block per batch, 256 threads.
// Point coords + running min-distances live in registers (strided
// ownership i = tid + 256*j, unrolled-index access only).
// Emit BEFORE update; argmax ties -> lowest global index.
// =====================================================================
__global__ __launch_bounds__(256)
void fps_kernel(const float* __restrict__ xyz,       // (B, N, 3)
                const int*   __restrict__ farthest0, // (B,)
                float*       __restrict__ sampled)   // (B, NPOINTS, 3) -> d_out head
{
    const int b   = blockIdx.x;
    const int tid = threadIdx.x;
    const float* base = xyz + (size_t)b * NPTS * 3;

    __shared__ float rmax[256];
    __shared__ int   rarg[256];

    float px[PPT], py[PPT], pz[PPT], pd[PPT];
#pragma unroll
    for (int j = 0; j < PPT; ++j) {
        const int i = tid + (j << 8);
        px[j] = base[i * 3 + 0];
        py[j] = base[i * 3 + 1];
        pz[j] = base[i * 3 + 2];
        pd[j] = 1e10f;
    }

    int far = farthest0[b];

    for (int it = 0; it < NPOINTS; ++it) {
        // centroid coords: uniform-address load (L2-resident broadcast)
        const float cx = base[far * 3 + 0];
        const float cy = base[far * 3 + 1];
        const float cz = base[far * 3 + 2];
        if (tid == 0) {
            float* o = sampled + ((size_t)b * NPOINTS + it) * 3;
            o[0] = cx; o[1] = cy; o[2] = cz;
        }

        float lmax = -1.0f;
        int   larg = 0;
#pragma unroll
        for (int j = 0; j < PPT; ++j) {
            const float dx = px[j] - cx;
            const float dy = py[j] - cy;
            const float dz = pz[j] - cz;
            const float d  = dx * dx + dy * dy + dz * dz;
            const float nd = fminf(pd[j], d);
            pd[j] = nd;
            if (nd > lmax) { lmax = nd; larg = tid + (j << 8); }  // ascending i -> first occurrence
        }
        rmax[tid] = lmax;
        rarg[tid] = larg;
        __syncthreads();
        for (int off = 128; off > 0; off >>= 1) {
            if (tid < off) {
                const float ov = rmax[tid + off];
                const int   oa = rarg[tid + off];
                if (ov > rmax[tid] || (ov == rmax[tid] && oa < rarg[tid])) {
                    rmax[tid] = ov; rarg[tid] = oa;
                }
            }
            __syncthreads();
        }
        far = rarg[0];
        __syncthreads();
    }
}

// =====================================================================
// Stage 2: kNN (32 nearest, nearest-first) + grouped-feature build.
// One block per (b, m) query.  Squared distances live in registers;
// selected entries invalidated with an unrolled compare (no dynamic
// register indexing).  Writes X0 laid out (Cin=512, P):
//   X0[m][(b*32+s)*67 + k],  k<3: xyz[j]-query, k>=3: points[b,j,k-3]
// =====================================================================
__global__ __launch_bounds__(256)
void knn_group_kernel(const float* __restrict__ xyz,     // (B,N,3)
                      const float* __restrict__ points,  // (B,N,C)
                      const float* __restrict__ sampled, // (B,NPOINTS,3)
                      float*       __restrict__ X0)      // (NPOINTS, PCOLS)
{
    const int gid = blockIdx.x;          // b*NPOINTS + m
    const int b   = gid >> 9;
    const int m   = gid & (NPOINTS - 1);
    const int tid = threadIdx.x;

    __shared__ int   sidx[NSAMP];
    __shared__ float rmin[256];
    __shared__ int   rarg[256];

    const float* base = xyz + (size_t)b * NPTS * 3;
    const float qx = sampled[(size_t)gid * 3 + 0];
    const float qy = sampled[(size_t)gid * 3 + 1];
    const float qz = sampled[(size_t)gid * 3 + 2];

    float pd[PPT];
#pragma unroll
    for (int j = 0; j < PPT; ++j) {
        const int i = tid + (j << 8);
        const float dx = base[i * 3 + 0] - qx;
        const float dy = base[i * 3 + 1] - qy;
        const float dz = base[i * 3 + 2] - qz;
        pd[j] = dx * dx + dy * dy + dz * dz;
    }

    for (int s = 0; s < NSAMP; ++s) {
        float lmin = FLT_MAX;
        int   larg = 0x7fffffff;
#pragma unroll
        for (int j = 0; j < PPT; ++j) {
            if (pd[j] < lmin) { lmin = pd[j]; larg = tid + (j << 8); }  // first occurrence
        }
        rmin[tid] = lmin;
        rarg[tid] = larg;
        __syncthreads();
        for (int off = 128; off > 0; off >>= 1) {
            if (tid < off) {
                const float ov = rmin[tid + off];
                const int   oa = rarg[tid + off];
                if (ov < rmin[tid] || (ov == rmin[tid] && oa < rarg[tid])) {
                    rmin[tid] = ov; rarg[tid] = oa;
                }
            }
            __syncthreads();
        }
        const int winner = rarg[0];
        if (tid == 0) sidx[s] = winner;
        // invalidate without dynamic register indexing
#pragma unroll
        for (int j = 0; j < PPT; ++j)
            if ((tid + (j << 8)) == winner) pd[j] = FLT_MAX;
        __syncthreads();
    }

    // emit grouped features for this (b, m)
    const float* pts = points + (size_t)b * NPTS * CFEAT;
    const float q[3] = {qx, qy, qz};
    for (int i = tid; i < NSAMP * KFEAT; i += 256) {
        const int s = i / KFEAT;
        const int k = i - s * KFEAT;
        const int j = sidx[s];
        float v;
        if (k < 3) v = base[j * 3 + k] - q[k];
        else       v = pts[(size_t)j * CFEAT + (k - 3)];
        X0[(size_t)m * PCOLS + (size_t)(b * NSAMP + s) * KFEAT + k] = v;
    }
}

// =====================================================================
// Stage 3: fp32 GEMM with V_WMMA_F32_16X16X4_F32.
//   Y[M,P] = W[M,K] @ f(X)[K,P] + bias[M]
// f(x) = identity          (BN=false, layer 1 input)
//      = relu(x*scale+sh)  (BN=true: producer's batchnorm fused into
//                           the B-tile staging load)
// Block tile 256x128, 8 waves (4 in M x 2 in N); each wave 64x64 =
// 4x4 accumulators.  B is stored TRANSPOSED in LDS so both A and B
// fragments are contiguous 8B-aligned pairs -> one ds_load_b64 each:
// 8 b64 loads (16 dwords) feed 16 WMMAs per k-step (1.0 dword/WMMA).
// Stride padding KC+4=20: fragment reads provably bank-conflict-free.
// =====================================================================
#define KC 16
#define LDP (KC + 4)      // 20-float padded stride (8B-aligned pairs, clean banks)
template <bool BN>
__global__ __launch_bounds__(256)
void gemm_wmma_f32(const float* __restrict__ W,
                   const float* __restrict__ X,
                   const float* __restrict__ bias,
                   const float* __restrict__ bscale,  // per-channel scale (BN)
                   const float* __restrict__ bshift,  // per-channel shift (BN)
                   float*       __restrict__ Y,
                   int K, int P)
{
    __shared__ float sA[256][LDP];       // [m][k]   20 KB
    __shared__ float sBt[128][LDP];      // [n][k]   10 KB (transposed)

    const int tid  = threadIdx.x;
    const int lane = tid & 31;
    const int wave = tid >> 5;           // 0..7
    const int wm   = wave & 3;           // M sub-tile base: wm*64
    const int wn   = wave >> 2;          // N sub-tile base: wn*64
    const int m0   = blockIdx.x * 256;
    const int n0   = blockIdx.y * 128;

    v8f acc[4][4] = {};

    const int mrow  = wm * 64 + (lane & 15);
    const int khalf = (lane >> 4) * 2;          // fragment k base per lane-half
    const int ncol  = wn * 64 + (lane & 15);

    for (int kc = 0; kc < K; kc += KC) {
        // stage A: 256 x 16 (16 elems/thread)
        for (int i = tid; i < 256 * KC; i += 256) {
            const int r = i >> 4, c = i & 15;
            sA[r][c] = W[(size_t)(m0 + r) * K + (kc + c)];
        }
        // stage B transposed: 16 x 128 -> sBt[n][k]
        // (optionally fusing BN+ReLU of the producer layer)
        for (int i = tid; i < KC * 128; i += 256) {
            const int r = i >> 7, c = i & 127;
            const int ch = kc + r;
            float v = X[(size_t)ch * P + (n0 + c)];
            if (BN) v = fmaxf(fmaf(v, bscale[ch], bshift[ch]), 0.0f);
            sBt[c][r] = v;
        }
        // prefetch next K-chunk while this one is consumed
        if (kc + KC < K) {
            __builtin_prefetch(&W[(size_t)(m0 + tid) * K + (kc + KC)], 0, 3);
            __builtin_prefetch(&X[(size_t)(kc + KC + (tid >> 4)) * P + n0 + ((tid & 15) << 3)], 0, 3);
        }
        __syncthreads();

#pragma unroll
        for (int k0 = 0; k0 < KC; k0 += 4) {
            v2f a[4], bb[4];
#pragma unroll
            for (int mi = 0; mi < 4; ++mi)
                a[mi] = *reinterpret_cast<const v2f*>(&sA[mrow + mi * 16][k0 + khalf]);
#pragma unroll
            for (int ni = 0; ni < 4; ++ni)
                bb[ni] = *reinterpret_cast<const v2f*>(&sBt[ncol + ni * 16][k0 + khalf]);
#pragma unroll
            for (int mi = 0; mi < 4; ++mi)
#pragma unroll
                for (int ni = 0; ni < 4; ++ni)
                    acc[mi][ni] = __builtin_amdgcn_wmma_f32_16x16x4_f32(
                        false, a[mi], false, bb[ni], (short)0, acc[mi][ni], false, false);
        }
        __syncthreads();
    }

    // epilogue: C/D layout -> reg r holds (m = r + 8*(lane>>4), n = lane&15)
    const int mh = (lane >> 4) * 8;
    const int nn = lane & 15;
#pragma unroll
    for (int mi = 0; mi < 4; ++mi) {
#pragma unroll
        for (int r = 0; r < 8; ++r) {
            const int gm = m0 + wm * 64 + mi * 16 + mh + r;
            const float bv = bias[gm];
            float* yrow = Y + (size_t)gm * P + n0 + wn * 64 + nn;
#pragma unroll
            for (int ni = 0; ni < 4; ++ni)
                yrow[ni * 16] = acc[mi][ni][r] + bv;
        }
    }
}

// =====================================================================
// Stage 4: BN batch statistics -> per-channel scale/shift.
//   scale = g * rsqrt(var + eps);  shift = beta - mean * scale
// =====================================================================
__global__ __launch_bounds__(256)
void bn_stats_kernel(const float* __restrict__ Y,
                     const float* __restrict__ g,
                     const float* __restrict__ beta,
                     float* __restrict__ scale,
                     float* __restrict__ shift,
                     int P)
{
    const int o   = blockIdx.x;
    const int tid = threadIdx.x;
    __shared__ float ssum[256];
    __shared__ float ssq[256];

    float s = 0.f, q = 0.f;
    const float* row = Y + (size_t)o * P;
    for (int i = tid; i < P; i += 256) {
        const float v = row[i];
        s += v;
        q += v * v;
    }
    ssum[tid] = s;
    ssq[tid]  = q;
    __syncthreads();
    for (int off = 128; off > 0; off >>= 1) {
        if (tid < off) { ssum[tid] += ssum[tid + off]; ssq[tid] += ssq[tid + off]; }
        __syncthreads();
    }
    if (tid == 0) {
        const float mu  = ssum[0] / (float)P;
        const float var = ssq[0] / (float)P - mu * mu;   // biased, matches jnp.var
        const float sc  = g[o] * rsqrtf(var + BNEPS);
        scale[o] = sc;
        shift[o] = beta[o] - mu * sc;
    }
}

// =====================================================================
// Stage 5: fused BN+ReLU (layer 3) + max over k(=67) -> (B, 1024, 32)
// =====================================================================
__global__ __launch_bounds__(256)
void bn_max_over_k_kernel(const float* __restrict__ Y3,
                          const float* __restrict__ scale,
                          const float* __restrict__ shift,
                          float* __restrict__ out_np)
{
    int i = blockIdx.x * 256 + threadIdx.x;          // i = b*32768 + o*32 + s
    const int total = BATCH * 1024 * NSAMP;
    if (i >= total) return;
    const int s = i & 31;
    const int o = (i >> 5) & 1023;
    const int b = i >> 15;
    const float sc = scale[o];
    const float sh = shift[o];
    const float* p = Y3 + (size_t)o * PCOLS + (size_t)(b * NSAMP + s) * KFEAT;
    float mx = fmaxf(fmaf(p[0], sc, sh), 0.0f);
#pragma unroll 1
    for (int k = 1; k < KFEAT; ++k) mx = fmaxf(mx, fmaxf(fmaf(p[k], sc, sh), 0.0f));
    out_np[i] = mx;
}

// =====================================================================
// Host launcher
// =====================================================================
extern "C" void kernel_launch(void* const* d_in, const int* in_sizes, int n_in,
                              void* d_out, int out_size, void* d_ws, size_t ws_size,
                              hipStream_t stream)
{
    (void)in_sizes; (void)n_in; (void)out_size; (void)ws_size;

    // inputs in setup_inputs() dict order
    const float* xyz    = (const float*)d_in[0];   // (16,8192,3)
    const float* points = (const float*)d_in[1];   // (16,8192,64)
    const int*   far0   = (const int*)  d_in[2];   // (16,)
    const float* w1 = (const float*)d_in[3];  const float* b1 = (const float*)d_in[4];
    const float* g1 = (const float*)d_in[5];  const float* be1 = (const float*)d_in[6];
    const float* w2 = (const float*)d_in[7];  const float* b2 = (const float*)d_in[8];
    const float* g2 = (const float*)d_in[9];  const float* be2 = (const float*)d_in[10];
    const float* w3 = (const float*)d_in[11]; const float* b3 = (const float*)d_in[12];
    const float* g3 = (const float*)d_in[13]; const float* be3 = (const float*)d_in[14];

    float* out_sampled = (float*)d_out;                              // (16,512,3)
    float* out_np      = (float*)d_out + BATCH * NPOINTS * 3;        // (16,1024,32)

    // workspace carve-up (256B aligned)
    char* ws = (char*)d_ws;
    size_t off = 0;
    auto carve = [&](size_t bytes) -> char* {
        char* p = ws + off;
        off += (bytes + 255) & ~(size_t)255;
        return p;
    };
    float* X0 = (float*)carve((size_t)512  * PCOLS * sizeof(float));  // grouped feats
    float* Y1 = (float*)carve((size_t)256  * PCOLS * sizeof(float));
    float* Y2 = (float*)carve((size_t)512  * PCOLS * sizeof(float));
    float* Y3 = (float*)carve((size_t)1024 * PCOLS * sizeof(float));
    float* bn_scale = (float*)carve(1024 * sizeof(float));
    float* bn_shift = (float*)carve(1024 * sizeof(float));

    // 1) FPS -> sampled_xyz (first output)
    fps_kernel<<<BATCH, 256, 0, stream>>>(xyz, far0, out_sampled);

    // 2) kNN + grouping -> X0 (512 x 34304)
    knn_group_kernel<<<BATCH * NPOINTS, 256, 0, stream>>>(xyz, points, out_sampled, X0);

    // 3) layer 1: Y1 = w1 @ X0 + b1
    gemm_wmma_f32<false><<<dim3(256 / 256, PCOLS / 128), 256, 0, stream>>>(
        w1, X0, b1, nullptr, nullptr, Y1, 512, PCOLS);
    bn_stats_kernel<<<256, 256, 0, stream>>>(Y1, g1, be1, bn_scale, bn_shift, PCOLS);

    // 4) layer 2: Y2 = w2 @ relu(bn(Y1)) + b2   (BN fused into B staging)
    gemm_wmma_f32<true><<<dim3(512 / 256, PCOLS / 128), 256, 0, stream>>>(
        w2, Y1, b2, bn_scale, bn_shift, Y2, 256, PCOLS);
    bn_stats_kernel<<<512, 256, 0, stream>>>(Y2, g2, be2, bn_scale, bn_shift, PCOLS);

    // 5) layer 3: Y3 = w3 @ relu(bn(Y2)) + b3
    gemm_wmma_f32<true><<<dim3(1024 / 256, PCOLS / 128), 256, 0, stream>>>(
        w3, Y2, b3, bn_scale, bn_shift, Y3, 512, PCOLS);
    bn_stats_kernel<<<1024, 256, 0, stream>>>(Y3, g3, be3, bn_scale, bn_shift, PCOLS);

    // 6) fused BN+ReLU + max over k -> new_points (second output)
    {
        const int total = BATCH * 1024 * NSAMP;
        bn_max_over_k_kernel<<<(total + 255) / 256, 256, 0, stream>>>(
            Y3, bn_scale, bn_shift, out_np);
    }
}